// CrossSSMFusion_69028714381787
// MI455X (gfx1250) — compile-verified
//
#include <hip/hip_runtime.h>
#include <hip/hip_bf16.h>
#include <cstddef>
#include <cstdint>

// ---------------------------------------------------------------------------
// CrossSSMFusion for MI455X (gfx1250, wave32).
//  * Main GEMMs: V_WMMA_F32_16X16X4_F32 fed from LDS; K-blocks of 32 are
//    bulk-staged with GLOBAL_LOAD_ASYNC_TO_LDS_B128 (ASYNCcnt), double
//    buffered: s_wait_asynccnt 6 overlaps next-tile DMA with current WMMAs.
//  * dt-projection (K=8) uses a small direct-load WMMA kernel.
//  * Selective scan is chunked (32 chunks of 128): phase1 per-chunk (P,Q)
//    summaries; phase2 32-step serial combine; phase3 replay + emit y.
//    B/C rows async-staged to LDS as well.
// ---------------------------------------------------------------------------

#define BSZ_   2
#define DIM_   128
#define L_     4096
#define BL_    8192          // BSZ_*L_
#define DI_    256           // d_inner
#define XZW_   512           // 2*d_inner
#define DTR_   8             // dt_rank
#define DS_    16            // d_state
#define XDBL_  40            // dt_rank + 2*d_state
#define CHUNK_ 128
#define NCH_   32            // L_/CHUNK_
#define LRS_   36            // LDS row stride (floats): 144B = 16B aligned, bank-spread

typedef float v2f __attribute__((ext_vector_type(2)));
typedef float v8f __attribute__((ext_vector_type(8)));

__device__ __forceinline__ float siluf(float x)     { return x / (1.f + expf(-x)); }
__device__ __forceinline__ float sigmoidf_(float x) { return 1.f / (1.f + expf(-x)); }
__device__ __forceinline__ float softplusf_(float x){ return x > 20.f ? x : log1pf(expf(x)); }

// gfx1250 VGLOBAL async global->LDS copies (tracked by ASYNCcnt).
__device__ __forceinline__ void async_g2l_b32(void* lds_ptr, const float* gptr)
{
    unsigned int lds_addr = (unsigned int)(uintptr_t)lds_ptr;  // low 32b of flat = LDS offset
    asm volatile("global_load_async_to_lds_b32 %0, %1, off"
                 :: "v"(lds_addr), "v"(gptr) : "memory");
}
__device__ __forceinline__ void async_g2l_b128(void* lds_ptr, const float* gptr)
{
    unsigned int lds_addr = (unsigned int)(uintptr_t)lds_ptr;
    asm volatile("global_load_async_to_lds_b128 %0, %1, off"
                 :: "v"(lds_addr), "v"(gptr) : "memory");
}
template<int N>
__device__ __forceinline__ void async_wait_le()
{
    asm volatile("s_wait_asynccnt %0" :: "i"(N) : "memory");
}

// ---------------------------------------------------------------------------
// LayerNorm over channels + transpose [B,C,N] -> [B*N, C]
// ---------------------------------------------------------------------------
__global__ void layernorm_tr(const float* __restrict__ feat,   // [B,C,N]
                             const float* __restrict__ g,
                             const float* __restrict__ b,
                             float* __restrict__ out)          // [B*N, C]
{
    int t  = blockIdx.x * blockDim.x + threadIdx.x;            // 0 .. B*N-1
    int bb = t >> 12;                                          // N = 4096
    int n  = t & (L_ - 1);
    const float* col = feat + (size_t)bb * DIM_ * L_ + n;
    float s = 0.f;
    for (int c = 0; c < DIM_; ++c) s += col[(size_t)c * L_];
    float m = s * (1.f / DIM_);
    float v = 0.f;
    for (int c = 0; c < DIM_; ++c) { float d = col[(size_t)c * L_] - m; v += d * d; }
    v *= (1.f / DIM_);
    float inv = rsqrtf(v + 1e-5f);
    float* orow = out + (size_t)t * DIM_;
    for (int c = 0; c < DIM_; ++c)
        orow[c] = (col[(size_t)c * L_] - m) * inv * g[c] + b[c];
}

// ---------------------------------------------------------------------------
// LDS-staged fp32 WMMA GEMM:  Out[m,n] = epi( sum_k A[m,k] * W[n,k] )
// Block = 256 threads (8 waves), tile 128x64, K-block 32, double-buffered.
// Per K-block each wave issues 6 async b128 DMAs (4 A-rows-worth + 2 W);
// while the next tile streams in, fragments come from LDS (ds_load_b64).
// Requires K % 32 == 0 and lda/ldw multiples of 4 floats.
// V_WMMA_F32_16X16X4_F32 fragment layouts:
//   A frag: lane m+16g -> {A[m][k+2g], A[m][k+2g+1]}
//   B frag: lane n+16g -> {W[n][k+2g], W[n][k+2g+1]}
//   D frag: vgpr i     -> row (i+8g), col = lane&15
// ---------------------------------------------------------------------------
template<int EPI, bool MASKED>
__global__ void wmma_gemm_lds(const float* __restrict__ A, int lda,
                              const float* __restrict__ W, int ldw,
                              float* __restrict__ Out, int ldo,
                              int K, int Nbound,
                              const float* __restrict__ bias)
{
    __shared__ float At[2][128 * LRS_];
    __shared__ float Wt[2][64 * LRS_];

    int tid  = threadIdx.x;
    int lane = tid & 31;
    int wave = tid >> 5;
    int g    = lane >> 4;
    int mr   = lane & 15;
    int g2   = 2 * g;
    int rowblk  = blockIdx.y * 128;
    int colbase = blockIdx.x << 6;

    float wmask[4];
#pragma unroll
    for (int tn = 0; tn < 4; ++tn) {
        int col = colbase + tn * 16 + mr;
        wmask[tn] = (!MASKED || col < Nbound) ? 1.0f : 0.0f;
    }

    auto load_tile = [&](int k0, int buf) {
#pragma unroll
        for (int j = 0; j < 4; ++j) {               // A: 128 rows x 32 cols
            int idx = tid + 256 * j;                // 0..1023
            int r = idx >> 3, cq = (idx & 7) << 2;  // 8 b128 chunks per row
            async_g2l_b128(&At[buf][r * LRS_ + cq],
                           A + (size_t)(rowblk + r) * lda + k0 + cq);
        }
#pragma unroll
        for (int j = 0; j < 2; ++j) {               // W: 64 rows x 32 cols
            int idx = tid + 256 * j;                // 0..511
            int r = idx >> 3, cq = (idx & 7) << 2;
            int wr = colbase + r;
            if (MASKED) wr = wr < Nbound ? wr : (Nbound - 1);
            async_g2l_b128(&Wt[buf][r * LRS_ + cq],
                           W + (size_t)wr * ldw + k0 + cq);
        }
    };

    v8f acc[4];
#pragma unroll
    for (int tn = 0; tn < 4; ++tn) acc[tn] = (v8f){0.f,0.f,0.f,0.f,0.f,0.f,0.f,0.f};

    int nkb = K >> 5;
    load_tile(0, 0);
    for (int kb = 0; kb < nkb; ++kb) {
        int buf = kb & 1;
        if (kb + 1 < nkb) {
            load_tile((kb + 1) << 5, buf ^ 1);   // next tile in flight
            async_wait_le<6>();                  // current tile's 6 DMAs done
        } else {
            async_wait_le<0>();
        }
        __syncthreads();

        const float* Ar = &At[buf][(wave * 16 + mr) * LRS_];
#pragma unroll
        for (int kk = 0; kk < 32; kk += 4) {
            int k = kk + g2;
            v2f a = *(const v2f*)(Ar + k);
#pragma unroll
            for (int tn = 0; tn < 4; ++tn) {
                v2f bf = *(const v2f*)(&Wt[buf][(tn * 16 + mr) * LRS_ + k]);
                if (MASKED) { bf.x *= wmask[tn]; bf.y *= wmask[tn]; }
                acc[tn] = __builtin_amdgcn_wmma_f32_16x16x4_f32(
                              false, a, false, bf, (short)0, acc[tn], false, false);
            }
        }
        __syncthreads();    // all waves done with buf before it is re-filled
    }

#pragma unroll
    for (int tn = 0; tn < 4; ++tn) {
        int col = colbase + tn * 16 + mr;
        if (!MASKED || col < Nbound) {
            float bv = (EPI == 1) ? bias[col] : 0.f;
#pragma unroll
            for (int i = 0; i < 8; ++i) {
                int row = rowblk + wave * 16 + i + 8 * g;
                float v = acc[tn][i];
                if (EPI == 1) v = softplusf_(v + bv);
                Out[(size_t)row * ldo + col] = v;
            }
        }
    }
}

// ---------------------------------------------------------------------------
// Direct-load WMMA GEMM for tiny K (dt projection, K=8).
// ---------------------------------------------------------------------------
template<int EPI>
__global__ void wmma_gemm_direct(const float* __restrict__ A, int lda,
                                 const float* __restrict__ W, int ldw,
                                 float* __restrict__ Out, int ldo,
                                 int K, int Nbound,
                                 const float* __restrict__ bias)
{
    int lane = threadIdx.x & 31;
    int wave = threadIdx.x >> 5;
    int g    = lane >> 4;
    int mr   = lane & 15;
    int row0    = (blockIdx.y * 8 + wave) << 4;
    int colbase = blockIdx.x << 6;

    const float* Arow = A + (size_t)(row0 + mr) * lda;
    const float* Wrow[4];
#pragma unroll
    for (int tn = 0; tn < 4; ++tn)
        Wrow[tn] = W + (size_t)(colbase + tn * 16 + mr) * ldw;

    v8f acc[4];
#pragma unroll
    for (int tn = 0; tn < 4; ++tn) acc[tn] = (v8f){0.f,0.f,0.f,0.f,0.f,0.f,0.f,0.f};

    for (int k0 = 0; k0 < K; k0 += 4) {
        int kk = k0 + 2 * g;
        v2f a = { Arow[kk], Arow[kk + 1] };
#pragma unroll
        for (int tn = 0; tn < 4; ++tn) {
            v2f bf = { Wrow[tn][kk], Wrow[tn][kk + 1] };
            acc[tn] = __builtin_amdgcn_wmma_f32_16x16x4_f32(
                          false, a, false, bf, (short)0, acc[tn], false, false);
        }
    }

#pragma unroll
    for (int tn = 0; tn < 4; ++tn) {
        int col = colbase + tn * 16 + mr;
        float bv = (EPI == 1) ? bias[col] : 0.f;
#pragma unroll
        for (int i = 0; i < 8; ++i) {
            int row = row0 + i + 8 * g;
            float v = acc[tn][i];
            if (EPI == 1) v = softplusf_(v + bv);
            Out[(size_t)row * ldo + col] = v;
        }
    }
}

// ---------------------------------------------------------------------------
// Depthwise causal conv (k=4) + SiLU, with optional sequence reversal.
// ---------------------------------------------------------------------------
__global__ void conv_silu(const float* __restrict__ xz,   // [B*L, 512] physical order
                          const float* __restrict__ cw,   // [256,4]
                          const float* __restrict__ cb,   // [256]
                          float* __restrict__ xconv,      // [B*L, 256] seq order
                          int rev)
{
    int d = threadIdx.x;            // channel
    int r = blockIdx.x;             // b*L + s   (sequence order)
    int b = r >> 12;
    int s = r & (L_ - 1);
    float acc = cb[d];
#pragma unroll
    for (int k = 0; k < 4; ++k) {
        int ss = s - 3 + k;
        if (ss >= 0) {
            int phys = (b << 12) + (rev ? (L_ - 1 - ss) : ss);
            acc += cw[d * 4 + k] * xz[(size_t)phys * XZW_ + d];
        }
    }
    xconv[(size_t)r * DI_ + d] = siluf(acc);
}

// ---------------------------------------------------------------------------
// Chunked selective scan, phase 1: per-chunk summaries
// ---------------------------------------------------------------------------
__global__ void scan_phase1(const float* __restrict__ delta,  // [B*L,256] seq order
                            const float* __restrict__ xconv,  // [B*L,256]
                            const float* __restrict__ xdbl,   // [B*L,40]  (B at +8)
                            const float* __restrict__ A_log,  // [256,16]
                            float* __restrict__ P,            // [B*NCH*256*16]
                            float* __restrict__ Q)
{
    __shared__ float Bs[CHUNK_ * DS_];
    int d   = threadIdx.x;
    int blk = blockIdx.x;                 // b*NCH + ch
    int b   = blk >> 5;
    int ch  = blk & 31;
    int base = (b << 12) + ch * CHUNK_;
    for (int i = threadIdx.x; i < CHUNK_ * DS_; i += 256) {
        int t = i >> 4, s = i & 15;
        async_g2l_b32(&Bs[i], &xdbl[(size_t)(base + t) * XDBL_ + 8 + s]);
    }
    async_wait_le<0>();
    __syncthreads();

    float Aa[DS_], h[DS_], p[DS_];
#pragma unroll
    for (int s = 0; s < DS_; ++s) {
        Aa[s] = -expf(A_log[d * DS_ + s]);
        h[s] = 0.f; p[s] = 1.f;
    }
    for (int t = 0; t < CHUNK_; ++t) {
        size_t rr = (size_t)(base + t) * DI_ + d;
        float de = delta[rr];
        float dx = de * xconv[rr];
#pragma unroll
        for (int s = 0; s < DS_; ++s) {
            float dA = expf(de * Aa[s]);
            h[s] = dA * h[s] + dx * Bs[t * DS_ + s];
            p[s] *= dA;
        }
    }
    size_t o = ((size_t)blk * DI_ + d) * DS_;
#pragma unroll
    for (int s = 0; s < DS_; ++s) { P[o + s] = p[s]; Q[o + s] = h[s]; }
}

// phase 2: serial combine across the 32 chunks
__global__ void scan_phase2(const float* __restrict__ P,
                            const float* __restrict__ Q,
                            float* __restrict__ Hin)
{
    int i  = blockIdx.x * blockDim.x + threadIdx.x;   // b*4096 + d*16 + s
    int b  = i >> 12;
    int ds = i & 4095;
    float h = 0.f;
    for (int ch = 0; ch < NCH_; ++ch) {
        size_t o = (size_t)(b * NCH_ + ch) * DI_ * DS_ + ds;
        Hin[o] = h;
        h = P[o] * h + Q[o];
    }
}

// phase 3: replay chunk from h_in, emit y = C.h + D*x  (seq order)
__global__ void scan_phase3(const float* __restrict__ delta,
                            const float* __restrict__ xconv,
                            const float* __restrict__ xdbl,
                            const float* __restrict__ A_log,
                            const float* __restrict__ Dp,     // [256]
                            const float* __restrict__ Hin,
                            float* __restrict__ y)            // [B*L,256] seq order
{
    __shared__ float Bs[CHUNK_ * DS_];
    __shared__ float Cs[CHUNK_ * DS_];
    int d   = threadIdx.x;
    int blk = blockIdx.x;
    int b   = blk >> 5;
    int ch  = blk & 31;
    int base = (b << 12) + ch * CHUNK_;
    for (int i = threadIdx.x; i < CHUNK_ * DS_; i += 256) {
        int t = i >> 4, s = i & 15;
        size_t ro = (size_t)(base + t) * XDBL_;
        async_g2l_b32(&Bs[i], &xdbl[ro + 8 + s]);
        async_g2l_b32(&Cs[i], &xdbl[ro + 24 + s]);
    }
    async_wait_le<0>();
    __syncthreads();

    float Aa[DS_], h[DS_];
    size_t ho = ((size_t)blk * DI_ + d) * DS_;
#pragma unroll
    for (int s = 0; s < DS_; ++s) {
        Aa[s] = -expf(A_log[d * DS_ + s]);
        h[s]  = Hin[ho + s];
    }
    float Dv = Dp[d];
    for (int t = 0; t < CHUNK_; ++t) {
        size_t rr = (size_t)(base + t) * DI_ + d;
        float de = delta[rr];
        float xv = xconv[rr];
        float dx = de * xv;
        float yv = 0.f;
#pragma unroll
        for (int s = 0; s < DS_; ++s) {
            float dA = expf(de * Aa[s]);
            h[s] = dA * h[s] + dx * Bs[t * DS_ + s];
            yv  += h[s] * Cs[t * DS_ + s];
        }
        y[rr] = yv + xv * Dv;
    }
}

// ---------------------------------------------------------------------------
// ysum[phys] (+)= y_seq[srow] * silu(z[phys])   (un-flips bwd branch)
// ---------------------------------------------------------------------------
__global__ void branch_accum(const float* __restrict__ y,     // seq order
                             const float* __restrict__ xz,    // physical order
                             float* __restrict__ ysum,        // physical order
                             int rev, int first)
{
    int d = threadIdx.x;
    int r = blockIdx.x;               // physical row b*L + t
    int b = r >> 12;
    int t = r & (L_ - 1);
    int srow = (b << 12) + (rev ? (L_ - 1 - t) : t);
    float z   = xz[(size_t)r * XZW_ + DI_ + d];
    float val = y[(size_t)srow * DI_ + d] * siluf(z);
    size_t o  = (size_t)r * DI_ + d;
    ysum[o] = first ? val : (ysum[o] + val);
}

// ---------------------------------------------------------------------------
// mean over L (two-stage), gate MLP, final cross-combine + transpose-out
// ---------------------------------------------------------------------------
__global__ void mean_part(const float* __restrict__ ssm, float* __restrict__ part)
{
    int c  = threadIdx.x;             // 0..127
    int gg = blockIdx.x;              // 16 groups of 256 rows
    int b  = blockIdx.y;
    float s = 0.f;
    int r0 = (b << 12) + gg * 256;
    for (int l = 0; l < 256; ++l) s += ssm[(size_t)(r0 + l) * DIM_ + c];
    part[(b * 16 + gg) * DIM_ + c] = s;
}

__global__ void mean_final(const float* __restrict__ part, float* __restrict__ mean)
{
    int c = threadIdx.x;
    int b = blockIdx.x;
    float s = 0.f;
    for (int gg = 0; gg < 16; ++gg) s += part[(b * 16 + gg) * DIM_ + c];
    mean[b * DIM_ + c] = s * (1.f / L_);
}

__global__ void gate_mlp(const float* __restrict__ mean,   // [B,128]
                         const float* __restrict__ W1,     // [16,128]
                         const float* __restrict__ b1,     // [16]
                         const float* __restrict__ W2,     // [128,16]
                         const float* __restrict__ b2,     // [128]
                         float* __restrict__ gOut)         // [B,128]
{
    __shared__ float h[16];
    int c = threadIdx.x;
    int b = blockIdx.x;
    const float* mrow = mean + b * DIM_;
    if (c < 16) {
        float s = b1[c];
        for (int k = 0; k < DIM_; ++k) s += W1[c * DIM_ + k] * mrow[k];
        h[c] = fmaxf(s, 0.f);
    }
    __syncthreads();
    float s = b2[c];
#pragma unroll
    for (int k = 0; k < 16; ++k) s += W2[c * 16 + k] * h[k];
    gOut[b * DIM_ + c] = sigmoidf_(s);
}

__global__ void final_combine(const float* __restrict__ assm,  // [B*L,128]
                              const float* __restrict__ bssm,
                              const float* __restrict__ ga,    // [B,128]
                              const float* __restrict__ gb,
                              const float* __restrict__ rs_p,
                              float* __restrict__ aout,        // [B,C,N]
                              float* __restrict__ bout)
{
    int gid = blockIdx.x * blockDim.x + threadIdx.x;   // over B*C*N
    int b   = gid >> 19;                               // C*N = 2^19
    int rem = gid & ((1 << 19) - 1);
    int c   = rem >> 12;
    int n   = rem & (L_ - 1);
    size_t rrow = ((size_t)(b << 12) + n) * DIM_ + c;
    float av = assm[rrow], bv = bssm[rrow];
    float gav = ga[b * DIM_ + c], gbv = gb[b * DIM_ + c];
    float rs = rs_p[0];
    aout[gid] = av * (1.f + gav) + bv * gav * rs;
    bout[gid] = bv * (1.f + gbv) + av * gbv * rs;
}

// ---------------------------------------------------------------------------
// Host orchestration.
// Param order = JAX pytree flatten (dict keys sorted alphabetically):
//   0 feat_a, 1 feat_b,
//   gate_a: 2 W1, 3 W2, 4 b1, 5 b2 ; gate_b: 6..9
//   mamba_a @10, mamba_b @33, each: [bwd(7), fwd(7), in_W, out_W, slc(7)]
//     branch fields: +0 A_log, +1 D, +2 conv_b, +3 conv_w, +4 dt_W, +5 dt_b, +6 xproj_W
//   norm_a: 56 b, 57 g ; norm_b: 58 b, 59 g ; 60 res_scale
// ---------------------------------------------------------------------------
extern "C" void kernel_launch(void* const* d_in, const int* in_sizes, int n_in,
                              void* d_out, int out_size, void* d_ws, size_t ws_size,
                              hipStream_t stream)
{
    (void)in_sizes; (void)n_in; (void)out_size; (void)ws_size;
    auto F = [&](int i) { return (const float*)d_in[i]; };

    float* ws = (float*)d_ws;
    size_t off = 0;
    auto alloc = [&](size_t n) { float* p = ws + off; off += n; return p; };

    float* xln0  = alloc((size_t)BL_ * DIM_);
    float* xln1  = alloc((size_t)BL_ * DIM_);
    float* xlnA[2] = { xln0, xln1 };
    float* xz    = alloc((size_t)BL_ * XZW_);
    float* xconv = alloc((size_t)BL_ * DI_);
    float* delta = alloc((size_t)BL_ * DI_);
    float* xdbl  = alloc((size_t)BL_ * XDBL_);
    float* ybr   = alloc((size_t)BL_ * DI_);
    float* Pb    = alloc((size_t)BSZ_ * NCH_ * DI_ * DS_);
    float* Qb    = alloc((size_t)BSZ_ * NCH_ * DI_ * DS_);
    float* Hin   = alloc((size_t)BSZ_ * NCH_ * DI_ * DS_);
    float* ysum  = alloc((size_t)BL_ * DI_);
    float* ssm0  = alloc((size_t)BL_ * DIM_);
    float* ssm1  = alloc((size_t)BL_ * DIM_);
    float* ssmA[2] = { ssm0, ssm1 };
    float* part  = alloc((size_t)BSZ_ * 16 * DIM_);
    float* mean0 = alloc((size_t)BSZ_ * DIM_);
    float* mean1 = alloc((size_t)BSZ_ * DIM_);
    float* meanA[2] = { mean0, mean1 };
    float* gA    = alloc((size_t)BSZ_ * DIM_);
    float* gB    = alloc((size_t)BSZ_ * DIM_);
    // total scratch ~72 MB (fits comfortably in L2's 192 MB)

    // LayerNorm + transpose for both streams
    layernorm_tr<<<BL_ / 256, 256, 0, stream>>>(F(0), F(57), F(56), xlnA[0]);
    layernorm_tr<<<BL_ / 256, 256, 0, stream>>>(F(1), F(59), F(58), xlnA[1]);

    const int broff[3] = { 7, 0, 16 };   // fwd, bwd, slc (NSLICES=1 -> slc order == fwd)
    const int brrev[3] = { 0, 1, 0 };

    for (int m = 0; m < 2; ++m) {
        int pb = (m == 0) ? 10 : 33;
        const float* in_W  = F(pb + 14);
        const float* out_W = F(pb + 15);

        // in-proj: [8192,128] x [128,512] -> xz   (LDS-staged, K=128)
        wmma_gemm_lds<0, false><<<dim3(XZW_ / 64, BL_ / 128), 256, 0, stream>>>(
            xlnA[m], DIM_, in_W, DIM_, xz, XZW_, DIM_, XZW_, nullptr);

        for (int br = 0; br < 3; ++br) {
            int bp  = pb + broff[br];
            int rev = brrev[br];
            conv_silu<<<BL_, 256, 0, stream>>>(xz, F(bp + 3), F(bp + 2), xconv, rev);
            // x-proj: [8192,256] x [256,40 (pad 64)] -> xdbl   (LDS-staged, masked)
            wmma_gemm_lds<0, true><<<dim3(1, BL_ / 128), 256, 0, stream>>>(
                xconv, DI_, F(bp + 6), DI_, xdbl, XDBL_, DI_, XDBL_, nullptr);
            // dt-proj + softplus: [8192,8] x [8,256] -> delta  (direct, K=8)
            wmma_gemm_direct<1><<<dim3(DI_ / 64, BL_ / 128), 256, 0, stream>>>(
                xdbl, XDBL_, F(bp + 4), DTR_, delta, DI_, DTR_, DI_, F(bp + 5));
            // chunked selective scan
            scan_phase1<<<BSZ_ * NCH_, 256, 0, stream>>>(delta, xconv, xdbl, F(bp + 0), Pb, Qb);
            scan_phase2<<<(BSZ_ * DI_ * DS_) / 256, 256, 0, stream>>>(Pb, Qb, Hin);
            scan_phase3<<<BSZ_ * NCH_, 256, 0, stream>>>(delta, xconv, xdbl, F(bp + 0),
                                                         F(bp + 1), Hin, ybr);
            branch_accum<<<BL_, 256, 0, stream>>>(ybr, xz, ysum, rev, br == 0 ? 1 : 0);
        }
        // out-proj: [8192,256] x [256,128] -> ssm  (LDS-staged, K=256)
        wmma_gemm_lds<0, false><<<dim3(DIM_ / 64, BL_ / 128), 256, 0, stream>>>(
            ysum, DI_, out_W, DI_, ssmA[m], DIM_, DI_, DIM_, nullptr);
        mean_part<<<dim3(16, BSZ_), 128, 0, stream>>>(ssmA[m], part);
        mean_final<<<BSZ_, 128, 0, stream>>>(part, meanA[m]);
    }

    // g_a = MLP(gate_a, mean(b_ssm)) ; g_b = MLP(gate_b, mean(a_ssm))
    gate_mlp<<<BSZ_, 128, 0, stream>>>(meanA[1], F(2), F(4), F(3), F(5), gA);
    gate_mlp<<<BSZ_, 128, 0, stream>>>(meanA[0], F(6), F(8), F(7), F(9), gB);

    float* aout = (float*)d_out;
    float* bout = (float*)d_out + (size_t)BSZ_ * DIM_ * L_;   // a_out is B*C*N elements
    final_combine<<<(BSZ_ * DIM_ * L_) / 256, 256, 0, stream>>>(
        ssmA[0], ssmA[1], gA, gB, F(60), aout, bout);
}